// MultiBoxLoss_34488587387300
// MI455X (gfx1250) — compile-verified
//
#include <hip/hip_runtime.h>
#include <hip/hip_bf16.h>

// ---------------------------------------------------------------------------
// MultiBoxLoss for MI455X (gfx1250, wave32).
// Streaming passes only (inputs ~48MB << 192MB L2, bandwidth-trivial at
// 23.3 TB/s). Hard-negative mining done with exact byte-wise radix select
// (O(P) passes) instead of the reference's argsort. Final scalar reductions
// use V_WMMA_F32_16X16X4_F32 as a ones-matrix reducer (64 f32 -> 16 rowsums
// per instruction, full f32 precision).
// ---------------------------------------------------------------------------

#define THRESH     0.35f
#define NEGPOS     7
#define VAR0       0.1f
#define VAR1       0.2f
#define VALID_GT   0.2f
#define MAXO       128

typedef float v2f_t __attribute__((ext_vector_type(2)));
typedef float v8f_t __attribute__((ext_vector_type(8)));

// ---------------------------------------------------------------------------
// Kernel 1: per-image SSD matching + per-prior stats. One block per image.
// ---------------------------------------------------------------------------
__global__ __launch_bounds__(256) void match_kernel(
    const float* __restrict__ loc_data,   // [B,P,4]
    const float* __restrict__ conf_data,  // [B,P,2]
    const float* __restrict__ priors,     // [P,4] center form
    const float* __restrict__ targets,    // [B,O,5]
    int P, int O,
    float* __restrict__ bt_ov,            // [B,P] best-truth overlap
    int*   __restrict__ bt_idx,           // [B,P] best-truth index
    float* __restrict__ lc_out,           // [B,P] mining loss
    int*   __restrict__ conf_t_out,       // [B,P] matched class
    float* __restrict__ loss_l_part,      // [Bp]
    int*   __restrict__ num_pos)          // [Bp]
{
  const int b    = blockIdx.x;
  const int tid  = threadIdx.x;
  const int nthr = blockDim.x;

  __shared__ float s_truth[MAXO * 4];
  __shared__ float s_label[MAXO];
  __shared__ unsigned long long s_bp[MAXO];   // (iou_bits<<32)|(P-1-p)
  __shared__ int   s_any;
  __shared__ float s_lsum;
  __shared__ int   s_pcnt;

  for (int o = tid; o < O; o += nthr) {
    const float* t = targets + ((size_t)b * O + o) * 5;
    s_truth[o * 4 + 0] = t[0];
    s_truth[o * 4 + 1] = t[1];
    s_truth[o * 4 + 2] = t[2];
    s_truth[o * 4 + 3] = t[3];
    s_label[o]         = t[4];
    s_bp[o]            = 0ull;
  }
  if (tid == 0) { s_any = 0; s_lsum = 0.f; s_pcnt = 0; }
  __syncthreads();

  // ---- Phase 1: IoU vs all truths; best truth per prior; best prior per truth
  for (int p = tid; p < P; p += nthr) {
    const float* pr = priors + (size_t)p * 4;
    float cx = pr[0], cy = pr[1], pw = pr[2], ph = pr[3];
    float px0 = cx - pw * 0.5f, py0 = cy - ph * 0.5f;
    float px1 = cx + pw * 0.5f, py1 = cy + ph * 0.5f;
    float area_p = (px1 - px0) * (py1 - py0);

    // warm L2 for phase 3 streaming reads (global_prefetch_b8)
    __builtin_prefetch(conf_data + ((size_t)b * P + p) * 2, 0, 1);
    __builtin_prefetch(loc_data  + ((size_t)b * P + p) * 4, 0, 1);

    float best = -1.f; int bo = 0;
    for (int o = 0; o < O; ++o) {
      float tx0 = s_truth[o * 4 + 0], ty0 = s_truth[o * 4 + 1];
      float tx1 = s_truth[o * 4 + 2], ty1 = s_truth[o * 4 + 3];
      float ix = fminf(tx1, px1) - fmaxf(tx0, px0);
      float iy = fminf(ty1, py1) - fmaxf(ty0, py0);
      ix = fmaxf(ix, 0.f); iy = fmaxf(iy, 0.f);
      float inter  = ix * iy;
      float area_t = (tx1 - tx0) * (ty1 - ty0);
      float iou = inter / (area_t + area_p - inter);
      if (iou > best) { best = iou; bo = o; }   // first-index tie-break
      unsigned long long packed =
          ((unsigned long long)__float_as_uint(iou) << 32) |
          (unsigned)(P - 1 - p);                // smaller p wins ties
      if (packed > s_bp[o]) atomicMax(&s_bp[o], packed);
    }
    bt_ov [(size_t)b * P + p] = best;
    bt_idx[(size_t)b * P + p] = bo;
  }
  __threadfence_block();
  __syncthreads();

  // ---- Phase 2: forced matches (sequential over O for scatter semantics)
  if (tid == 0) {
    int any = 0;
    for (int o = 0; o < O; ++o) {
      unsigned long long pk = s_bp[o];
      float ov = __uint_as_float((unsigned)(pk >> 32));
      int   pb = P - 1 - (int)(unsigned)(pk & 0xFFFFFFFFull);
      int valid = (ov >= VALID_GT) ? 1 : 0;
      any |= valid;
      bt_idx[(size_t)b * P + pb] = o;           // idx overwrite for ALL o
      if (valid) bt_ov[(size_t)b * P + pb] = 2.0f;
    }
    s_any = any;
  }
  __threadfence_block();
  __syncthreads();
  const int any_valid = s_any;

  // ---- Phase 3: conf_t, encode, smooth-L1, mining loss
  float lsum = 0.f; int pcnt = 0;
  for (int p = tid; p < P; p += nthr) {
    float ov = bt_ov [(size_t)b * P + p];
    int   o  = bt_idx[(size_t)b * P + p];
    int conf = (int)s_label[o];
    if (ov < THRESH)  conf = 0;
    if (!any_valid)   conf = 0;
    int pos = conf > 0;

    float mx0 = s_truth[o * 4 + 0], my0 = s_truth[o * 4 + 1];
    float mx1 = s_truth[o * 4 + 2], my1 = s_truth[o * 4 + 3];
    const float* pr = priors + (size_t)p * 4;
    float cxp = pr[0], cyp = pr[1], pw = pr[2], ph = pr[3];
    float gcx = ((mx0 + mx1) * 0.5f - cxp) / (VAR0 * pw);
    float gcy = ((my0 + my1) * 0.5f - cyp) / (VAR0 * ph);
    float gw  = __logf((mx1 - mx0) / pw) / VAR1;
    float gh  = __logf((my1 - my0) / ph) / VAR1;
    if (!any_valid) { gcx = 0.f; gcy = 0.f; gw = 0.f; gh = 0.f; }

    if (pos) {
      const float* ld = loc_data + ((size_t)b * P + p) * 4;
      float lt[4] = {gcx, gcy, gw, gh};
      #pragma unroll
      for (int k = 0; k < 4; ++k) {
        float d = ld[k] - lt[k];
        float a = fabsf(d);
        lsum += (a < 1.f) ? 0.5f * d * d : (a - 0.5f);
      }
      pcnt++;
    }

    const float* cd = conf_data + ((size_t)b * P + p) * 2;
    float c0 = cd[0], c1 = cd[1];
    float m   = fmaxf(c0, c1);
    float lse = m + __logf(__expf(c0 - m) + __expf(c1 - m));
    float g   = conf ? c1 : c0;
    float lc  = pos ? 0.f : (lse - g);
    lc_out    [(size_t)b * P + p] = lc;
    conf_t_out[(size_t)b * P + p] = conf;
  }
  atomicAdd(&s_lsum, lsum);
  atomicAdd(&s_pcnt, pcnt);
  __syncthreads();
  if (tid == 0) { loss_l_part[b] = s_lsum; num_pos[b] = s_pcnt; }
}

// ---------------------------------------------------------------------------
// Kernel 2: per-image radix-select top-k on lc + stable tie scan + CE sum.
// ---------------------------------------------------------------------------
__global__ __launch_bounds__(256) void select_kernel(
    const float* __restrict__ lc,         // [B,P] (>= 0 always)
    const int*   __restrict__ conf_t,     // [B,P]
    const float* __restrict__ conf_data,  // [B,P,2]
    int P,
    const int*   __restrict__ num_pos,    // [Bp]
    float* __restrict__ loss_c_part)      // [Bp]
{
  const int b    = blockIdx.x;
  const int tid  = threadIdx.x;
  const int nthr = blockDim.x;
  const int lane = tid & 31;
  const int wv   = tid >> 5;
  const int nw   = nthr >> 5;

  __shared__ int      s_hist[256];
  __shared__ unsigned s_prefix;
  __shared__ int      s_want;
  __shared__ int      s_wcnt[32];
  __shared__ int      s_running;
  __shared__ float    s_csum;

  int k = num_pos[b] * NEGPOS;          // num_neg = min(7*num_pos, P-1)
  if (k > P - 1) k = P - 1;

  unsigned T = 0xFFFFFFFFu;             // threshold key (k-th largest)
  int r = 0;                            // ties of T to keep (stable order)
  if (k > 0) {                          // k is block-uniform
    if (tid == 0) { s_prefix = 0u; s_want = k; }
    __syncthreads();
    for (int byte = 3; byte >= 0; --byte) {
      for (int i = tid; i < 256; i += nthr) s_hist[i] = 0;
      __syncthreads();
      unsigned prefix = s_prefix;
      unsigned hmask  = (byte == 3) ? 0u : (0xFFFFFFFFu << ((byte + 1) * 8));
      for (int p = tid; p < P; p += nthr) {
        unsigned key = __float_as_uint(lc[(size_t)b * P + p]);
        if ((key & hmask) == (prefix & hmask))
          atomicAdd(&s_hist[(key >> (byte * 8)) & 0xFF], 1);
      }
      __syncthreads();
      if (tid == 0) {
        int want = s_want, c = 0, bin;
        for (bin = 255; bin >= 0; --bin) {
          if (c + s_hist[bin] >= want) break;
          c += s_hist[bin];
        }
        if (bin < 0) bin = 0;
        s_want   = want - c;
        s_prefix = prefix | ((unsigned)bin << (byte * 8));
      }
      __syncthreads();
    }
    T = s_prefix; r = s_want;
  }

  // ---- Ordered tie scan (stable-sort semantics) + selected CE sum
  if (tid == 0) { s_running = 0; s_csum = 0.f; }
  __syncthreads();

  float csum = 0.f;
  for (int base = 0; base < P; base += nthr) {
    int p = base + tid;
    int validp = (p < P);
    unsigned key = 0; int pos = 0, gt = 0, tie = 0;
    float ce = 0.f;
    if (validp) {
      key = __float_as_uint(lc[(size_t)b * P + p]);
      int c = conf_t[(size_t)b * P + p];
      pos = c > 0;
      gt  = (k > 0) && (key > T);
      tie = (k > 0) && (key == T);
      const float* cd = conf_data + ((size_t)b * P + p) * 2;
      float c0 = cd[0], c1 = cd[1];
      float m   = fmaxf(c0, c1);
      float lse = m + __logf(__expf(c0 - m) + __expf(c1 - m));
      ce = lse - (c ? c1 : c0);
    }
    unsigned long long bal = __ballot(tie);
    if (lane == 0) s_wcnt[wv] = (int)__popcll(bal);
    __syncthreads();
    if (tid == 0) {
      int acc = s_running;
      for (int w = 0; w < nw; ++w) { int t = s_wcnt[w]; s_wcnt[w] = acc; acc += t; }
      s_running = acc;
    }
    __syncthreads();
    int myrank = s_wcnt[wv] + (int)__popcll(bal & ((1ull << lane) - 1ull));
    int neg = gt || (tie && (myrank < r));
    if (validp && (pos || neg)) csum += ce;
    __syncthreads();   // protect s_wcnt/s_running for next chunk
  }
  atomicAdd(&s_csum, csum);
  __syncthreads();
  if (tid == 0) loss_c_part[b] = s_csum;
}

// ---------------------------------------------------------------------------
// Kernel 3: WMMA-based final reduction. One wave32.
// A (16x4 f32, 64 partials in 2 VGPRs) x ones(4x16) -> D rows = row-sums.
// Lanes 0..15 hold rowsums 0..7 in acc[0..7]; lanes 16..31 hold rowsums 8..15.
// ---------------------------------------------------------------------------
__global__ __launch_bounds__(32) void finalize_kernel(
    const float* __restrict__ loss_l_part,
    const float* __restrict__ loss_c_part,
    const int*   __restrict__ num_pos,
    int Bp,                                  // multiple of 64, zero-padded
    float* __restrict__ out)                 // out[0]=loss_l/N, out[1]=loss_c/N
{
  const int lane = threadIdx.x;
  v2f_t ones; ones[0] = 1.f; ones[1] = 1.f;
  v8f_t accL = {}, accC = {}, accN = {};

  for (int base = 0; base < Bp; base += 64) {
    v2f_t a, c, n;
    a[0] = loss_l_part[base + 2 * lane];
    a[1] = loss_l_part[base + 2 * lane + 1];
    c[0] = loss_c_part[base + 2 * lane];
    c[1] = loss_c_part[base + 2 * lane + 1];
    n[0] = (float)num_pos[base + 2 * lane];
    n[1] = (float)num_pos[base + 2 * lane + 1];
    accL = __builtin_amdgcn_wmma_f32_16x16x4_f32(false, a, false, ones,
                                                 (short)0, accL, false, false);
    accC = __builtin_amdgcn_wmma_f32_16x16x4_f32(false, c, false, ones,
                                                 (short)0, accC, false, false);
    accN = __builtin_amdgcn_wmma_f32_16x16x4_f32(false, n, false, ones,
                                                 (short)0, accN, false, false);
  }

  float tL = 0.f, tC = 0.f, tN = 0.f;
  #pragma unroll
  for (int i = 0; i < 8; ++i) { tL += accL[i]; tC += accC[i]; tN += accN[i]; }
  const int other = lane ^ 16;               // combine the two row-halves
  tL += __shfl(tL, other);
  tC += __shfl(tC, other);
  tN += __shfl(tN, other);

  if (lane == 0) {
    float N = fmaxf(tN, 1.f);
    out[0] = tL / N;
    out[1] = tC / N;
  }
}

// ---------------------------------------------------------------------------
// Host launcher
// ---------------------------------------------------------------------------
extern "C" void kernel_launch(void* const* d_in, const int* in_sizes, int n_in,
                              void* d_out, int out_size, void* d_ws, size_t ws_size,
                              hipStream_t stream) {
  const float* loc_data  = (const float*)d_in[0];
  const float* conf_data = (const float*)d_in[1];
  const float* priors    = (const float*)d_in[2];
  const float* targets   = (const float*)d_in[3];

  const int P  = in_sizes[2] / 4;
  const int B  = in_sizes[0] / (P * 4);
  const int O  = in_sizes[3] / (B * 5);
  const int Bp = ((B + 63) / 64) * 64;

  // Carve workspace (256B aligned chunks).
  char* w = (char*)d_ws;
  auto carve = [&](size_t bytes) -> void* {
    void* p = (void*)w;
    w += ((bytes + 255) / 256) * 256;
    return p;
  };
  float* bt_ov       = (float*)carve((size_t)B * P * sizeof(float));
  int*   bt_idx      = (int*)  carve((size_t)B * P * sizeof(int));
  float* lc          = (float*)carve((size_t)B * P * sizeof(float));
  int*   conf_t      = (int*)  carve((size_t)B * P * sizeof(int));
  float* loss_l_part = (float*)carve((size_t)Bp * sizeof(float));
  float* loss_c_part = (float*)carve((size_t)Bp * sizeof(float));
  int*   num_pos     = (int*)  carve((size_t)Bp * sizeof(int));

  // Zero the padded partial arrays (graph-capturable memset nodes).
  hipMemsetAsync(loss_l_part, 0, (size_t)Bp * sizeof(float), stream);
  hipMemsetAsync(loss_c_part, 0, (size_t)Bp * sizeof(float), stream);
  hipMemsetAsync(num_pos,     0, (size_t)Bp * sizeof(int),   stream);

  match_kernel<<<B, 256, 0, stream>>>(loc_data, conf_data, priors, targets,
                                      P, O, bt_ov, bt_idx, lc, conf_t,
                                      loss_l_part, num_pos);
  select_kernel<<<B, 256, 0, stream>>>(lc, conf_t, conf_data, P,
                                       num_pos, loss_c_part);
  finalize_kernel<<<1, 32, 0, stream>>>(loss_l_part, loss_c_part, num_pos,
                                        Bp, (float*)d_out);
}